// IDNN_15178414424665
// MI455X (gfx1250) — compile-verified
//
#include <hip/hip_runtime.h>
#include <math.h>

// ---------------------------------------------------------------------------
// Free-energy MLP + grad + Hessian via forward-mode channels, fp32 WMMA.
//   B=524288, D=4, T=2, H=64.  15 channels/sample: 1 fwd + 4 tangent + 10 pair.
//   Each hidden layer = GEMM (15*16 x 64) @ (64 x 64) per 16-sample tile,
//   done with V_WMMA_F32_16X16X4_F32, data staged in LDS.
//   B (weight) fragments are hoisted to registers per n-tile and reused
//   across the wave's channels (4x fewer LDS reads in the GEMM hot loop).
// ---------------------------------------------------------------------------

typedef __attribute__((ext_vector_type(2))) float v2f;
typedef __attribute__((ext_vector_type(8))) float v8f;

#define BATCH   524288
#define HID     64
#define NCHAN   15
#define BLOCK   128     // 4 waves of 32
#define TILE_M  16

// LDS float offsets (total 40672 floats = 162,688 bytes -> 2 WGs / 320KB WGP)
#define W1o 0        // 6*64
#define B1o 384      // 64
#define W2o 448      // 64*64
#define B2o 4544     // 64
#define W3o 4608     // 64*64
#define B3o 8704     // 64
#define WOo 8768     // 64
#define S1o 8832     // 16*64 sigmoid buffer
#define XSo 9856     // 16*6 input tile
#define CHA 9952     // 15 * 16*64
#define CHB 25312    // 15 * 16*64
#define LDS_FLOATS 40672

static __device__ __forceinline__ float softplus_f(float z) {
    return fmaxf(z, 0.0f) + log1pf(expf(-fabsf(z)));
}
static __device__ __forceinline__ float sigmoid_f(float z) {
    return 1.0f / (1.0f + expf(-z));
}

// One hidden layer: chanOut[c] = elementwise( chanIn[c] @ W + (bias for fwd) )
static __device__ __forceinline__ void layer_step(
    float* lds, int chanIn, int chanOut, int Wo, int Bo,
    int tid, int wave, int lane)
{
    const int mrow = lane & 15;
    const int half = lane >> 4;          // 0: lanes 0-15, 1: lanes 16-31

    // ---- GEMM phase: channels split across the 4 waves (4,4,4,3) ----------
    const int c0 = wave * 4;
    const int c1 = (c0 + 4 < NCHAN) ? c0 + 4 : NCHAN;
    #pragma unroll
    for (int nt = 0; nt < 4; ++nt) {
        // Hoist the 16 B (weight) fragments for this n-tile into registers:
        // W row k, col n at Wo + k*64 + n; lane covers n = nt*16 + mrow,
        // k-halves offset by 2*half (ISA 4x16 f32 B layout).
        const float* Bb = &lds[Wo + (2 * half) * HID + nt * 16 + mrow];
        v2f bf[16];
        #pragma unroll
        for (int kb = 0; kb < 16; ++kb) {
            bf[kb].x = Bb[kb * 4 * HID];
            bf[kb].y = Bb[kb * 4 * HID + HID];
        }
        for (int c = c0; c < c1; ++c) {
            // A fragment base: element (m, k) at chanIn + m*64 + k;
            // lane gets m = mrow, k-halves offset by 2*half.
            const float* Ab = &lds[chanIn + c * (TILE_M * HID) + mrow * HID + 2 * half];
            v8f acc = {};
            #pragma unroll
            for (int kb = 0; kb < 16; ++kb) {
                v2f a = *(const v2f*)(Ab + kb * 4);          // ds_load_b64
                acc = __builtin_amdgcn_wmma_f32_16x16x4_f32(
                    /*neg_a=*/false, a, /*neg_b=*/false, bf[kb],
                    /*c_mod=*/(short)0, acc,
                    /*reuse_a=*/false, /*reuse_b=*/false);
            }
            // C/D layout: VGPR r holds M = r + 8*half, N = nt*16 + mrow
            float* Ob = &lds[chanOut + c * (TILE_M * HID) + (8 * half) * HID + nt * 16 + mrow];
            #pragma unroll
            for (int r = 0; r < 8; ++r) Ob[r * HID] = acc[r];
        }
    }
    __syncthreads();

    // ---- Phase 2: forward channel nonlinearity + sigma buffer -------------
    for (int e = tid; e < TILE_M * HID; e += BLOCK) {
        const int n = e & (HID - 1);
        const float z  = lds[chanOut + e] + lds[Bo + n];
        lds[S1o + e]   = sigmoid_f(z);
        lds[chanOut + e] = softplus_f(z);
    }
    __syncthreads();

    // ---- Phase 3a: pair channels (read RAW tangent Z before overwrite) ----
    {
        int p = 0;
        #pragma unroll
        for (int i = 0; i < 4; ++i) {
            #pragma unroll
            for (int j = i; j < 4; ++j, ++p) {
                for (int r = tid; r < TILE_M * HID; r += BLOCK) {
                    const float s1 = lds[S1o + r];
                    const float s2 = s1 * (1.0f - s1);
                    const float zi = lds[chanOut + (1 + i) * (TILE_M * HID) + r];
                    const float zj = lds[chanOut + (1 + j) * (TILE_M * HID) + r];
                    const float zu = lds[chanOut + (5 + p) * (TILE_M * HID) + r];
                    lds[chanOut + (5 + p) * (TILE_M * HID) + r] = s2 * zi * zj + s1 * zu;
                }
            }
        }
    }
    __syncthreads();

    // ---- Phase 3b: tangent channels ---------------------------------------
    for (int e = tid; e < 4 * TILE_M * HID; e += BLOCK) {
        const int i = e >> 10;           // TILE_M*HID == 1024
        const int r = e & 1023;
        lds[chanOut + (1 + i) * (TILE_M * HID) + r] *= lds[S1o + r];
    }
    __syncthreads();
}

__global__ __launch_bounds__(BLOCK)
void idnn_fe_hess_kernel(
    const float* __restrict__ x1, const float* __restrict__ x4,
    const float* __restrict__ W1, const float* __restrict__ b1,
    const float* __restrict__ W2, const float* __restrict__ b2,
    const float* __restrict__ W3, const float* __restrict__ b3,
    const float* __restrict__ Wout, float* __restrict__ out)
{
    extern __shared__ float lds[];
    const int tid  = threadIdx.x;
    const int wave = tid >> 5;
    const int lane = tid & 31;
    const int g0   = blockIdx.x * TILE_M;

    // ---- cooperative loads: weights + input tile --------------------------
    for (int i = tid; i < 6 * HID; i += BLOCK) lds[W1o + i] = W1[i];
    for (int i = tid; i < HID * HID; i += BLOCK) {
        lds[W2o + i] = W2[i];
        lds[W3o + i] = W3[i];
    }
    if (tid < HID) {
        lds[B1o + tid] = b1[tid];
        lds[B2o + tid] = b2[tid];
        lds[B3o + tid] = b3[tid];
        lds[WOo + tid] = Wout[tid];
    }
    if (tid < 64) {                                  // x1 tile: 16 x 4
        const int m = tid >> 2, d = tid & 3;
        lds[XSo + m * 6 + d] = x1[(size_t)(g0 + m) * 4 + d];
    } else if (tid < 96) {                           // x4 tile: 16 x 2
        const int t2 = tid - 64, m = t2 >> 1, d = t2 & 1;
        lds[XSo + m * 6 + 4 + d] = x4[(size_t)(g0 + m) * 2 + d];
    }
    __syncthreads();

    // ---- Layer 1 (K=6, scalar): seed all 15 channels into CHA -------------
    for (int e = tid; e < TILE_M * HID; e += BLOCK) {
        const int m = e >> 6, n = e & (HID - 1);
        float z = lds[B1o + n];
        float w1n[4];
        #pragma unroll
        for (int d = 0; d < 4; ++d) {
            const float w = lds[W1o + d * HID + n];
            w1n[d] = w;
            z += lds[XSo + m * 6 + d] * w;
        }
        z += lds[XSo + m * 6 + 4] * lds[W1o + 4 * HID + n];
        z += lds[XSo + m * 6 + 5] * lds[W1o + 5 * HID + n];
        const float s1 = sigmoid_f(z);
        const float s2 = s1 * (1.0f - s1);
        lds[CHA + e] = softplus_f(z);
        #pragma unroll
        for (int i = 0; i < 4; ++i)
            lds[CHA + (1 + i) * (TILE_M * HID) + e] = s1 * w1n[i];
        int p = 0;
        #pragma unroll
        for (int i = 0; i < 4; ++i)
            #pragma unroll
            for (int j = i; j < 4; ++j, ++p)
                lds[CHA + (5 + p) * (TILE_M * HID) + e] = s2 * w1n[i] * w1n[j];
    }
    __syncthreads();

    // ---- Hidden layers 2 and 3 (WMMA GEMMs) -------------------------------
    layer_step(lds, CHA, CHB, W2o, B2o, tid, wave, lane);   // CHA -> CHB
    layer_step(lds, CHB, CHA, W3o, B3o, tid, wave, lane);   // CHB -> CHA

    // ---- Output projection: 15 channels x 16 samples dot Wout -------------
    for (int e = tid; e < NCHAN * TILE_M; e += BLOCK) {
        const int c = e >> 4, m = e & 15;
        float acc = 0.0f;
        #pragma unroll 8
        for (int n = 0; n < HID; ++n)
            acc += lds[CHA + c * (TILE_M * HID) + m * HID + n] * lds[WOo + n];
        const size_t g = (size_t)(g0 + m);
        if (c == 0) {
            out[g] = acc;                                       // y (B,1)
        } else if (c < 5) {
            out[(size_t)BATCH + g * 4 + (c - 1)] = acc;         // dy (B,4)
        } else {
            const int p = c - 5;                                // Hessian pair
            const int i = (p < 4) ? 0 : (p < 7) ? 1 : (p < 9) ? 2 : 3;
            const int start = (i == 0) ? 0 : (i == 1) ? 4 : (i == 2) ? 7 : 9;
            const int j = i + (p - start);
            float* ddy = out + (size_t)BATCH * 5 + g * 16;
            ddy[i * 4 + j] = acc;
            ddy[j * 4 + i] = acc;                               // symmetric
        }
    }
}

extern "C" void kernel_launch(void* const* d_in, const int* in_sizes, int n_in,
                              void* d_out, int out_size, void* d_ws, size_t ws_size,
                              hipStream_t stream) {
    (void)in_sizes; (void)n_in; (void)d_ws; (void)ws_size; (void)out_size;
    const float* x1   = (const float*)d_in[0];
    const float* x4   = (const float*)d_in[1];
    const float* W1   = (const float*)d_in[2];
    const float* b1   = (const float*)d_in[3];
    const float* W2   = (const float*)d_in[4];
    const float* b2   = (const float*)d_in[5];
    const float* W3   = (const float*)d_in[6];
    const float* b3   = (const float*)d_in[7];
    const float* Wout = (const float*)d_in[8];
    float* out = (float*)d_out;

    const int grid = BATCH / TILE_M;                 // 32768 tiles
    const size_t shmem = (size_t)LDS_FLOATS * sizeof(float);   // 162,688 B
    idnn_fe_hess_kernel<<<dim3(grid), dim3(BLOCK), shmem, stream>>>(
        x1, x4, W1, b1, W2, b2, W3, b3, Wout, out);
}